// GraphGPS_38457137168573
// MI455X (gfx1250) — compile-verified
//
#include <hip/hip_runtime.h>
#include <cstdint>
#include <cstddef>

// ---------------- static dims (match reference) ----------------
#define N_NODES 16384
#define CH      128
#define NEDGE   262144
#define NB      32
#define SS      512
#define PEDIM   16
#define NHEAD   4
#define DH      32
#define OUTD    3
#define EMBD    112      // C - PE
#define VEPS    1e-5f

typedef float v2f __attribute__((ext_vector_type(2)));
typedef float v8f __attribute__((ext_vector_type(8)));

__device__ __forceinline__ v8f wmma4(v8f acc, v2f a, v2f b) {
  // D = A(16x4) * B(4x16) + C, fp32
  return __builtin_amdgcn_wmma_f32_16x16x4_f32(false, a, false, b, (short)0, acc,
                                               false, false);
}

// =====================================================================
// Generic GEMM: Y[M,Cout] = X[M,K] @ W[Cout,K]^T + bias, optional ReLU.
// One wave per 16x16 output tile, K stepped by 4 with v_wmma_f32_16x16x4_f32.
// =====================================================================
__global__ void gemm_xwT(const float* __restrict__ X, const float* __restrict__ W,
                         const float* __restrict__ bias, float* __restrict__ Y,
                         int M, int K, int Cout, int fuse_relu)
{
  int gid  = blockIdx.x * blockDim.x + threadIdx.x;
  int wave = gid >> 5;
  int lane = gid & 31;
  int tilesN = Cout >> 4;
  int total  = (M >> 4) * tilesN;
  if (wave >= total) return;
  int tm = wave / tilesN, tn = wave - tm * tilesN;
  int r    = lane & 15;
  int koff = (lane >> 4) << 1;

  const float* xrow = X + (size_t)(tm * 16 + r) * K;
  const float* wrow = W + (size_t)(tn * 16 + r) * K;

  v8f acc = {0.f, 0.f, 0.f, 0.f, 0.f, 0.f, 0.f, 0.f};
  for (int k = 0; k < K; k += 4) {
    v2f a = *(const v2f*)(xrow + k + koff);
    v2f b = *(const v2f*)(wrow + k + koff);
    acc = wmma4(acc, a, b);
  }

  int n  = tn * 16 + r;
  int mb = tm * 16 + ((lane >> 4) << 3);
  float bv = bias ? bias[n] : 0.0f;
#pragma unroll
  for (int i = 0; i < 8; ++i) {
    float v = acc[i] + bv;
    if (fuse_relu) v = fmaxf(v, 0.0f);
    Y[(size_t)(mb + i) * Cout + n] = v;
  }
}

// =====================================================================
// Edge message GEMM: m[e,:] = pre_w @ [h[dst[e]] ; h[src[e]]] + pre_b
// A rows gathered per edge; K = 256 split into the two 128-halves.
// =====================================================================
__global__ void gemm_edge_pre(const float* __restrict__ h, const int* __restrict__ esrc,
                              const int* __restrict__ edst, const float* __restrict__ W,
                              const float* __restrict__ bias, float* __restrict__ Mout)
{
  int gid  = blockIdx.x * blockDim.x + threadIdx.x;
  int wave = gid >> 5;
  int lane = gid & 31;
  const int tilesN = CH / 16;                 // 8
  int total = (NEDGE / 16) * tilesN;
  if (wave >= total) return;
  int tm = wave / tilesN, tn = wave - tm * tilesN;
  int r    = lane & 15;
  int koff = (lane >> 4) << 1;

  int e = tm * 16 + r;
  const float* hd   = h + (size_t)edst[e] * CH;
  const float* hs   = h + (size_t)esrc[e] * CH;
  const float* wrow = W + (size_t)(tn * 16 + r) * (2 * CH);

  __builtin_prefetch(hs, 0, 0);   // global_prefetch_b8: warm the src half

  v8f acc = {0.f, 0.f, 0.f, 0.f, 0.f, 0.f, 0.f, 0.f};
#pragma unroll 4
  for (int k = 0; k < CH; k += 4) {
    v2f a = *(const v2f*)(hd + k + koff);
    v2f b = *(const v2f*)(wrow + k + koff);
    acc = wmma4(acc, a, b);
  }
#pragma unroll 4
  for (int k = 0; k < CH; k += 4) {
    v2f a = *(const v2f*)(hs + k + koff);
    v2f b = *(const v2f*)(wrow + CH + k + koff);
    acc = wmma4(acc, a, b);
  }

  int n  = tn * 16 + r;
  int mb = tm * 16 + ((lane >> 4) << 3);
  float bv = bias[n];
#pragma unroll
  for (int i = 0; i < 8; ++i)
    Mout[(size_t)(mb + i) * CH + n] = acc[i] + bv;
}

// =====================================================================
// Attention scores: sc[bh, q, k] = (Q . K) / sqrt(dh)   (K-dim = 32)
// qkv layout: [N, 3*CH], Q at +hh*32, K at +128+hh*32, V at +256+hh*32
// =====================================================================
__global__ void attn_scores(const float* __restrict__ qkv, float* __restrict__ sc)
{
  int gid  = blockIdx.x * blockDim.x + threadIdx.x;
  int wave = gid >> 5;
  int lane = gid & 31;
  int total = NB * NHEAD * 32 * 32;
  if (wave >= total) return;
  int tk = wave & 31, tq = (wave >> 5) & 31, bh = wave >> 10;
  int b = bh >> 2, hh = bh & 3;
  int r = lane & 15, koff = (lane >> 4) << 1;

  const float* qrow = qkv + (size_t)(b * SS + tq * 16 + r) * (3 * CH) + hh * DH;
  const float* krow = qkv + (size_t)(b * SS + tk * 16 + r) * (3 * CH) + CH + hh * DH;

  v8f acc = {0.f, 0.f, 0.f, 0.f, 0.f, 0.f, 0.f, 0.f};
#pragma unroll
  for (int k = 0; k < DH; k += 4) {
    v2f a  = *(const v2f*)(qrow + k + koff);
    v2f bb = *(const v2f*)(krow + k + koff);
    acc = wmma4(acc, a, bb);
  }

  const float scale = 0.17677669529663687f;   // 1/sqrt(32)
  int n  = tk * 16 + r;
  int mb = tq * 16 + ((lane >> 4) << 3);
  float* base = sc + (size_t)bh * SS * SS;
#pragma unroll
  for (int i = 0; i < 8; ++i)
    base[(size_t)(mb + i) * SS + n] = acc[i] * scale;
}

// Row softmax, one wave per row of length 512.
__global__ void softmax_rows(float* __restrict__ sc)
{
  int gid  = blockIdx.x * blockDim.x + threadIdx.x;
  int wave = gid >> 5;
  int lane = gid & 31;
  if (wave >= NB * NHEAD * SS) return;
  float* row = sc + (size_t)wave * SS;

  float mx = -3.4e38f;
  for (int i = lane; i < SS; i += 32) mx = fmaxf(mx, row[i]);
#pragma unroll
  for (int off = 16; off > 0; off >>= 1) mx = fmaxf(mx, __shfl_xor(mx, off));

  float s = 0.f;
  for (int i = lane; i < SS; i += 32) { float e = __expf(row[i] - mx); row[i] = e; s += e; }
#pragma unroll
  for (int off = 16; off > 0; off >>= 1) s += __shfl_xor(s, off);

  float inv = 1.0f / s;
  for (int i = lane; i < SS; i += 32) row[i] *= inv;
}

// A.V: o[b*S+q, hh*32 + td*16 + n] = sum_k sc[bh,q,k] * V[k,n]   (K = 512)
__global__ void attn_av(const float* __restrict__ sc, const float* __restrict__ qkv,
                        float* __restrict__ o)
{
  int gid  = blockIdx.x * blockDim.x + threadIdx.x;
  int wave = gid >> 5;
  int lane = gid & 31;
  int total = NB * NHEAD * 32 * 2;
  if (wave >= total) return;
  int td = wave & 1, tq = (wave >> 1) & 31, bh = wave >> 6;
  int b = bh >> 2, hh = bh & 3;
  int r = lane & 15, koff = (lane >> 4) << 1;

  const float* srow = sc + (size_t)bh * SS * SS + (size_t)(tq * 16 + r) * SS;
  const float* vcol = qkv + (size_t)(b * SS) * (3 * CH) + 2 * CH + hh * DH + td * 16 + r;

  v8f acc = {0.f, 0.f, 0.f, 0.f, 0.f, 0.f, 0.f, 0.f};
  for (int k = 0; k < SS; k += 4) {
    v2f a  = *(const v2f*)(srow + k + koff);
    v2f bb = { vcol[(size_t)(k + koff) * (3 * CH)],
               vcol[(size_t)(k + koff + 1) * (3 * CH)] };
    acc = wmma4(acc, a, bb);
  }

  int n  = td * 16 + r;
  int mb = tq * 16 + ((lane >> 4) << 3);
#pragma unroll
  for (int i = 0; i < 8; ++i)
    o[(size_t)(b * SS + mb + i) * CH + hh * DH + n] = acc[i];
}

// =====================================================================
// CSR construction (dst-sorted adjacency), built once per launch.
// =====================================================================
__global__ void zero_int(int* __restrict__ p, int n)
{
  int i = blockIdx.x * blockDim.x + threadIdx.x;
  if (i < n) p[i] = 0;
}

__global__ void count_dst(const int* __restrict__ dst, int* __restrict__ cnt)
{
  int e = blockIdx.x * blockDim.x + threadIdx.x;
  if (e < NEDGE) atomicAdd(&cnt[dst[e]], 1);
}

// single block of 1024; cnt_cursor is read (counts) then overwritten (cursors)
__global__ void build_rowptr(int* __restrict__ cnt_cursor, int* __restrict__ rowptr)
{
  __shared__ int sdata[1024];
  int t = threadIdx.x;
  int loc[16];
  int s = 0;
#pragma unroll
  for (int j = 0; j < 16; ++j) { loc[j] = s; s += cnt_cursor[t * 16 + j]; }
  sdata[t] = s;
  __syncthreads();
  for (int off = 1; off < 1024; off <<= 1) {
    int v = (t >= off) ? sdata[t - off] : 0;
    __syncthreads();
    sdata[t] += v;
    __syncthreads();
  }
  int base = (t == 0) ? 0 : sdata[t - 1];
#pragma unroll
  for (int j = 0; j < 16; ++j) {
    int rp = base + loc[j];
    rowptr[t * 16 + j]     = rp;
    cnt_cursor[t * 16 + j] = rp;   // cursor copy for scatter
  }
  if (t == 1023) rowptr[N_NODES] = sdata[1023];
}

__global__ void scatter_edges(const int* __restrict__ dst, int* __restrict__ cursor,
                              int* __restrict__ eid)
{
  int e = blockIdx.x * blockDim.x + threadIdx.x;
  if (e >= NEDGE) return;
  int pos = atomicAdd(&cursor[dst[e]], 1);
  eid[pos] = e;
}

// =====================================================================
// PNA aggregation: one wave per node; edges visited in ascending edge-id
// order (deterministic fp sums). agg[n] = [x | mean | max | std], width 512.
// =====================================================================
__global__ void pna_aggregate(const float* __restrict__ h, const float* __restrict__ Mmsg,
                              const int* __restrict__ rowptr, const int* __restrict__ eid,
                              float* __restrict__ agg)
{
  int gid  = blockIdx.x * blockDim.x + threadIdx.x;
  int wave = gid >> 5;
  int lane = gid & 31;
  if (wave >= N_NODES) return;
  int beg = rowptr[wave], end = rowptr[wave + 1];
  int deg = end - beg;

  float s0[4] = {0.f, 0.f, 0.f, 0.f};
  float s1[4] = {0.f, 0.f, 0.f, 0.f};
  float mx[4];
#pragma unroll
  for (int j = 0; j < 4; ++j) mx[j] = -3.4e38f;

  int prev = -1;
  for (int it = 0; it < deg; ++it) {
    int cur = 0x7fffffff;
    for (int p = beg + lane; p < end; p += 32) {
      int id = eid[p];
      if (id > prev && id < cur) cur = id;
    }
#pragma unroll
    for (int off = 16; off > 0; off >>= 1) {
      int other = __shfl_xor(cur, off);
      cur = min(cur, other);
    }
    prev = cur;
    const float* mr = Mmsg + (size_t)cur * CH;
#pragma unroll
    for (int j = 0; j < 4; ++j) {
      float v = mr[lane + 32 * j];
      s0[j] += v; s1[j] += v * v; mx[j] = fmaxf(mx[j], v);
    }
  }

  float denom = (float)(deg > 0 ? deg : 1);
  float*       arow = agg + (size_t)wave * (4 * CH);
  const float* hrow = h + (size_t)wave * CH;
#pragma unroll
  for (int j = 0; j < 4; ++j) {
    int c = lane + 32 * j;
    float mean  = s0[j] / denom;
    float mean2 = s1[j] / denom;
    float var   = mean2 - mean * mean;
    if (var < 0.f) var = 0.f;
    arow[c]          = hrow[c];
    arow[CH + c]     = mean;
    arow[2 * CH + c] = (deg > 0) ? mx[j] : 0.0f;
    arow[3 * CH + c] = sqrtf(var + 1e-5f);
  }
}

// =====================================================================
// BatchNorm (training stats, biased var) with optional fused residual.
// =====================================================================
__global__ void bn_stats(const float* __restrict__ a, const float* __restrict__ b,
                         float* __restrict__ stats, int M, int C)
{
  __shared__ float ss[256];
  __shared__ float sq[256];
  int c = blockIdx.x, t = threadIdx.x;
  float s = 0.f, s2 = 0.f;
  for (int i = t; i < M; i += blockDim.x) {
    float v = a[(size_t)i * C + c];
    if (b) v += b[(size_t)i * C + c];
    s += v; s2 += v * v;
  }
  ss[t] = s; sq[t] = s2;
  __syncthreads();
  for (int o = 128; o > 0; o >>= 1) {
    if (t < o) { ss[t] += ss[t + o]; sq[t] += sq[t + o]; }
    __syncthreads();
  }
  if (t == 0) {
    float mean = ss[0] / (float)M;
    stats[2 * c]     = mean;
    stats[2 * c + 1] = sq[0] / (float)M - mean * mean;
  }
}

__global__ void bn_apply(const float* __restrict__ a, const float* __restrict__ b,
                         const float* __restrict__ stats, const float* __restrict__ g,
                         const float* __restrict__ be, float* __restrict__ out,
                         int M, int C)
{
  int idx = blockIdx.x * blockDim.x + threadIdx.x;
  if (idx >= M * C) return;
  int c = idx % C;
  float v = a[idx];
  if (b) v += b[idx];
  out[idx] = (v - stats[2 * c]) * rsqrtf(stats[2 * c + 1] + VEPS) * g[c] + be[c];
}

__global__ void add_vec(const float* __restrict__ a, const float* __restrict__ b,
                        float* __restrict__ o, int n)
{
  int i = blockIdx.x * blockDim.x + threadIdx.x;
  if (i < n) o[i] = a[i] + b[i];
}

// =====================================================================
// Input embedding: h = [node_emb[x] | pe_lin(pe_bn(pe))]
// =====================================================================
__global__ void embed_kernel(const int* __restrict__ xid, const float* __restrict__ pe,
                             const float* __restrict__ stats, const float* __restrict__ g,
                             const float* __restrict__ be, const float* __restrict__ pw,
                             const float* __restrict__ pb, const float* __restrict__ emb,
                             float* __restrict__ h)
{
  int idx = blockIdx.x * blockDim.x + threadIdx.x;
  if (idx >= N_NODES * CH) return;
  int nrow = idx / CH, c = idx - nrow * CH;
  float v;
  if (c < EMBD) {
    v = emb[(size_t)xid[nrow] * EMBD + c];
  } else {
    int j = c - EMBD;
    float acc = pb[j];
#pragma unroll
    for (int k = 0; k < PEDIM; ++k) {
      float pn = (pe[nrow * PEDIM + k] - stats[2 * k]) *
                 rsqrtf(stats[2 * k + 1] + VEPS) * g[k] + be[k];
      acc += pw[j * PEDIM + k] * pn;
    }
    v = acc;
  }
  h[idx] = v;
}

// Per-graph max pool: graphs are contiguous 512-row segments.
__global__ void pool_max(const float* __restrict__ h, float* __restrict__ pooled)
{
  int b = blockIdx.x;
  int c = threadIdx.x;          // blockDim == 128
  float m = -3.4e38f;
  const float* base = h + (size_t)b * SS * CH + c;
  for (int s = 0; s < SS; ++s) m = fmaxf(m, base[(size_t)s * CH]);
  pooled[b * CH + c] = m;
}

// Tiny MLP head in one block: 128 -> 64 -> 32 -> 3 with ReLU between.
__global__ void mlp_head(const float* __restrict__ pooled,
                         const float* __restrict__ w0, const float* __restrict__ b0,
                         const float* __restrict__ w1, const float* __restrict__ b1,
                         const float* __restrict__ w2, const float* __restrict__ b2,
                         float* __restrict__ out)
{
  __shared__ float bufA[NB * CH];
  __shared__ float bufB[NB * 64];
  __shared__ float bufC[NB * 32];
  int t = threadIdx.x;
  for (int i = t; i < NB * CH; i += blockDim.x) bufA[i] = pooled[i];
  __syncthreads();
  for (int i = t; i < NB * 64; i += blockDim.x) {
    int rr = i >> 6, cc = i & 63;
    float a = b0[cc];
    for (int k = 0; k < CH; ++k) a += w0[cc * CH + k] * bufA[rr * CH + k];
    bufB[i] = fmaxf(a, 0.f);
  }
  __syncthreads();
  for (int i = t; i < NB * 32; i += blockDim.x) {
    int rr = i >> 5, cc = i & 31;
    float a = b1[cc];
    for (int k = 0; k < 64; ++k) a += w1[cc * 64 + k] * bufB[rr * 64 + k];
    bufC[i] = fmaxf(a, 0.f);
  }
  __syncthreads();
  for (int i = t; i < NB * OUTD; i += blockDim.x) {
    int rr = i / OUTD, cc = i - rr * OUTD;
    float a = b2[cc];
    for (int k = 0; k < 32; ++k) a += w2[cc * 32 + k] * bufC[rr * 32 + k];
    out[i] = a;
  }
}

// =====================================================================
// Host-side orchestration
// =====================================================================
extern "C" void kernel_launch(void* const* d_in, const int* in_sizes, int n_in,
                              void* d_out, int out_size, void* d_ws, size_t ws_size,
                              hipStream_t stream)
{
  (void)in_sizes; (void)n_in; (void)out_size; (void)ws_size;

  const int*   x     = (const int*)d_in[0];
  const float* pe    = (const float*)d_in[1];
  const int*   ei    = (const int*)d_in[2];
  const int*   batch = (const int*)d_in[3]; (void)batch;   // contiguous segments
  const float* emb   = (const float*)d_in[4];
  const float* pe_w  = (const float*)d_in[5];
  const float* pe_b  = (const float*)d_in[6];
  const float* pbn_g = (const float*)d_in[7];
  const float* pbn_b = (const float*)d_in[8];

  // layers: 20 tensors each, dict insertion order
  const float* LW[2][20];
  for (int l = 0; l < 2; ++l)
    for (int i = 0; i < 20; ++i)
      LW[l][i] = (const float*)d_in[9 + l * 20 + i];
  const float* mlp_w0 = (const float*)d_in[49];
  const float* mlp_b0 = (const float*)d_in[50];
  const float* mlp_w1 = (const float*)d_in[51];
  const float* mlp_b1 = (const float*)d_in[52];
  const float* mlp_w2 = (const float*)d_in[53];
  const float* mlp_b2 = (const float*)d_in[54];

  const int* esrc = ei;
  const int* edst = ei + NEDGE;

  // ---- workspace layout ----
  float* wsp    = (float*)d_ws;
  float* hbuf   = wsp; wsp += (size_t)N_NODES * CH;
  float* ta     = wsp; wsp += (size_t)N_NODES * CH;
  float* tb     = wsp; wsp += (size_t)N_NODES * CH;
  float* tc     = wsp; wsp += (size_t)N_NODES * CH;
  float* stats  = wsp; wsp += 1024;
  float* tmp1   = wsp; wsp += (size_t)N_NODES * 4 * CH;   // agg / qkv / ffn hidden
  float* big    = wsp; wsp += (size_t)NEDGE * CH;         // == NB*NHEAD*SS*SS floats
  float* pooled = wsp; wsp += NB * CH;
  int*   icnt   = (int*)wsp;                               // cursor after scan
  int*   irow   = icnt + N_NODES;
  int*   ieid   = irow + (N_NODES + 1);

  auto gemm = [&](const float* X, const float* W, const float* bias, float* Y,
                  int M, int K, int Cout, int relu) {
    int waves  = (M / 16) * (Cout / 16);
    int blocks = (waves * 32 + 255) / 256;
    gemm_xwT<<<blocks, 256, 0, stream>>>(X, W, bias, Y, M, K, Cout, relu);
  };

  const int NC      = N_NODES * CH;
  const int NCBLK   = (NC + 255) / 256;

  // ---- CSR by dst (edges identical across layers: build once) ----
  zero_int<<<(N_NODES + 255) / 256, 256, 0, stream>>>(icnt, N_NODES);
  count_dst<<<NEDGE / 256, 256, 0, stream>>>(edst, icnt);
  build_rowptr<<<1, 1024, 0, stream>>>(icnt, irow);
  scatter_edges<<<NEDGE / 256, 256, 0, stream>>>(edst, icnt, ieid);

  // ---- input embedding ----
  bn_stats<<<PEDIM, 256, 0, stream>>>(pe, nullptr, stats, N_NODES, PEDIM);
  embed_kernel<<<NCBLK, 256, 0, stream>>>(x, pe, stats, pbn_g, pbn_b, pe_w, pe_b,
                                          emb, hbuf);

  // ---- GPS layers ----
  for (int l = 0; l < 2; ++l) {
    const float* const* P = LW[l];

    // PNA local branch
    {
      int waves = (NEDGE / 16) * (CH / 16);
      gemm_edge_pre<<<(waves * 32 + 255) / 256, 256, 0, stream>>>(
          hbuf, esrc, edst, P[0], P[1], big);
    }
    pna_aggregate<<<(N_NODES * 32 + 255) / 256, 256, 0, stream>>>(hbuf, big, irow,
                                                                  ieid, tmp1);
    gemm(tmp1, P[2], P[3], ta, N_NODES, 4 * CH, CH, 0);   // post
    gemm(ta,   P[4], P[5], tb, N_NODES, CH, CH, 0);       // lin
    bn_stats<<<CH, 256, 0, stream>>>(tb, hbuf, stats, N_NODES, CH);
    bn_apply<<<NCBLK, 256, 0, stream>>>(tb, hbuf, stats, P[14], P[15], ta,
                                        N_NODES, CH);     // h_local -> ta

    // global attention branch
    gemm(hbuf, P[6], P[7], tmp1, N_NODES, CH, 3 * CH, 0); // qkv
    {
      int waves = NB * NHEAD * 32 * 32;
      attn_scores<<<(waves * 32 + 255) / 256, 256, 0, stream>>>(tmp1, big);
    }
    softmax_rows<<<(NB * NHEAD * SS * 32) / 256, 256, 0, stream>>>(big);
    {
      int waves = NB * NHEAD * 32 * 2;
      attn_av<<<(waves * 32 + 255) / 256, 256, 0, stream>>>(big, tmp1, tb);
    }
    gemm(tb, P[8], P[9], tc, N_NODES, CH, CH, 0);         // out proj
    bn_stats<<<CH, 256, 0, stream>>>(tc, hbuf, stats, N_NODES, CH);
    bn_apply<<<NCBLK, 256, 0, stream>>>(tc, hbuf, stats, P[16], P[17], tb,
                                        N_NODES, CH);     // h_attn -> tb

    // combine + FFN
    add_vec<<<NCBLK, 256, 0, stream>>>(ta, tb, tc, NC);   // out -> tc
    gemm(tc,   P[10], P[11], tmp1, N_NODES, CH, 2 * CH, 1);   // m1 + relu
    gemm(tmp1, P[12], P[13], tb,   N_NODES, 2 * CH, CH, 0);   // m2
    bn_stats<<<CH, 256, 0, stream>>>(tc, tb, stats, N_NODES, CH);
    bn_apply<<<NCBLK, 256, 0, stream>>>(tc, tb, stats, P[18], P[19], hbuf,
                                        N_NODES, CH);     // new h
  }

  // ---- pooling + MLP head ----
  pool_max<<<NB, CH, 0, stream>>>(hbuf, pooled);
  mlp_head<<<1, 256, 0, stream>>>(pooled, mlp_w0, mlp_b0, mlp_w1, mlp_b1,
                                  mlp_w2, mlp_b2, (float*)d_out);
}